// CombinedPolyLoss_64458869178587
// MI455X (gfx1250) — compile-verified
//
#include <hip/hip_runtime.h>
#include <hip/hip_bf16.h>

typedef float v2f __attribute__((ext_vector_type(2)));
typedef float v8f __attribute__((ext_vector_type(8)));

#define EPS_CLAMP  1.0e-4f
#define POLY_EPS   2.0f
#define CLS_LMDA   0.05f
#define NBLK       1024
#define NTHR       256

#if defined(__gfx1250__) && __has_builtin(__builtin_amdgcn_wmma_f32_16x16x4_f32)
#define USE_WMMA_REDUCE 1
#endif

// Full 32-lane wave sum, broadcast to all lanes.
// Uses two V_WMMA_F32_16X16X4_F32 (f32 in, f32 acc -> exact fp32 FMA tree):
//  WMMA1: A[m][0]=s(m), A[m][2]=s(m+16), rest 0; B=ones -> D[m][n]=s(m)+s(m+16)
//  WMMA2: fold D's 8 regs into A2 (even/odd) -> every D2 element = full sum.
__device__ __forceinline__ float wave_sum32(float s) {
#ifdef USE_WMMA_REDUCE
    v2f ones = {1.0f, 1.0f};
    v2f a    = {s, 0.0f};
    v8f c    = {};
    v8f d  = __builtin_amdgcn_wmma_f32_16x16x4_f32(false, a,  false, ones,
                                                   (short)0, c, false, false);
    v2f a2 = { d[0] + d[2] + d[4] + d[6],
               d[1] + d[3] + d[5] + d[7] };
    v8f d2 = __builtin_amdgcn_wmma_f32_16x16x4_f32(false, a2, false, ones,
                                                   (short)0, c, false, false);
    return d2[0];
#else
    for (int off = 16; off > 0; off >>= 1) s += __shfl_xor(s, off, 32);
    return s;
#endif
}

// Valid only on threadIdx.x == 0. Deterministic (fixed order, no atomics).
__device__ __forceinline__ float block_sum(float v, float* lds) {
    float w = wave_sum32(v);
    const int wave = threadIdx.x >> 5;
    const int lane = threadIdx.x & 31;
    __syncthreads();                 // protect lds reuse across calls
    if (lane == 0) lds[wave] = w;
    __syncthreads();
    float r = 0.0f;
    if (threadIdx.x == 0) {
        const int nw = blockDim.x >> 5;
        for (int i = 0; i < nw; ++i) r += lds[i];
    }
    return r;
}

// Poly-BCE per element; t is exactly 0.0 or 1.0.
// p clamped to [1e-4, 1-1e-4] => log(pt) in [-9.22, ~0], the -100 clamp in the
// reference can never bind, so one log suffices: poly = -log(pt) + 2*(1-pt).
__device__ __forceinline__ float poly_elem(float x, float t) {
    float p = 1.0f / (1.0f + __expf(-x));          // v_exp_f32 + v_rcp_f32
    p = fminf(fmaxf(p, EPS_CLAMP), 1.0f - EPS_CLAMP);
    float pt = (t > 0.0f) ? p : 1.0f - p;
    float lg = fmaxf(__logf(pt), -100.0f);         // v_log_f32
    return POLY_EPS * (1.0f - pt) - lg;
}

// Pass 1: stream both arrays with float4 loads (global_load_b128),
// grid-stride (uniform 9 iters/thread for this size), per-block partial sums.
__global__ __launch_bounds__(NTHR)
void poly_partials_kernel(const float* __restrict__ hm,
                          const float* __restrict__ tg,
                          float* __restrict__ partial, int n4) {
    const int tid    = blockIdx.x * blockDim.x + threadIdx.x;
    const int stride = gridDim.x * blockDim.x;
    const float4* __restrict__ hv = (const float4*)hm;
    const float4* __restrict__ tv = (const float4*)tg;

    float acc = 0.0f;
    for (int i = tid; i < n4; i += stride) {
        float4 x = hv[i];
        float4 t = tv[i];
        acc += poly_elem(x.x, t.x);
        acc += poly_elem(x.y, t.y);
        acc += poly_elem(x.z, t.z);
        acc += poly_elem(x.w, t.w);
    }

    __shared__ float lds[NTHR / 32];
    float bs = block_sum(acc, lds);
    if (threadIdx.x == 0) partial[blockIdx.x] = bs;
}

// Pass 2: one block folds the 1024 partials -> hm_loss, and does the
// 64-element classifier BCE -> cls_loss. Fully convergent control flow.
__global__ __launch_bounds__(NTHR)
void finalize_kernel(const float* __restrict__ partial, int npart,
                     const float* __restrict__ cls_p,
                     const float* __restrict__ cls_t, int ncls,
                     float* __restrict__ out, float hm_scale) {
    const int tid = threadIdx.x;
    __shared__ float lds[NTHR / 32];

    // heatmap partial fold (npart = 1024 = 4 * NTHR)
    float acc = 0.0f;
    for (int i = tid; i < npart; i += NTHR) acc += partial[i];
    float hm_sum = block_sum(acc, lds);

    // classifier BCE: masked, loads always in-bounds, no divergent branch
    const int  ci    = tid & (64 - 1);
    const bool valid = (tid < ncls);
    float p = cls_p[ci];
    float t = cls_t[ci];
    float logp   = fmaxf(__logf(p),        -100.0f);
    float log1mp = fmaxf(__logf(1.0f - p), -100.0f);
    float bce = -(t * logp + (1.0f - t) * log1mp);
    float cls_sum = block_sum(valid ? bce : 0.0f, lds);

    if (tid == 0) {
        out[0] = hm_sum * hm_scale;                       // sum/(H*W)/B
        out[1] = (cls_sum / (float)ncls) * CLS_LMDA;      // mean * lambda
    }
}

extern "C" void kernel_launch(void* const* d_in, const int* in_sizes, int n_in,
                              void* d_out, int out_size, void* d_ws, size_t ws_size,
                              hipStream_t stream) {
    const float* hm     = (const float*)d_in[0];   // (64,1,384,384) f32
    const float* tg     = (const float*)d_in[1];   // (64,384,384)   f32 (same flat layout)
    const float* cls_p  = (const float*)d_in[2];   // (64,1)
    const float* cls_t  = (const float*)d_in[3];   // (64,1)
    float*       out    = (float*)d_out;           // [hm_loss, cls_loss]
    float*       part   = (float*)d_ws;            // NBLK floats scratch

    const int n   = in_sizes[0];                   // 9,437,184
    const int n4  = n / 4;                         // exact (no tail)
    const int B   = 64, HW = 384 * 384;
    const float hm_scale = 1.0f / ((float)HW * (float)B);

    poly_partials_kernel<<<NBLK, NTHR, 0, stream>>>(hm, tg, part, n4);
    finalize_kernel<<<1, NTHR, 0, stream>>>(part, NBLK, cls_p, cls_t,
                                            in_sizes[2], out, hm_scale);
}